// DFTD3ModuleStatic_43147241456154
// MI455X (gfx1250) — compile-verified
//
#include <hip/hip_runtime.h>
#include <hip/hip_bf16.h>

typedef __attribute__((ext_vector_type(2))) float v2f;
typedef __attribute__((ext_vector_type(8))) float v8f;

#define INV_AUTOANG (1.0f/0.52917726f)
#define AUTOEV 27.21138505f
#define K1F 16.0f
#define K3F (-4.0f)
#define CUTOFF2 (95.0f*95.0f)
#define CNTHR2  (40.0f*40.0f)
#define RS6 1.217f
#define S18 0.722f

#define NATOM 512
#define NSH   27
#define NCOL  (NATOM*NSH)     /* 13824 columns: n = s*512 + j */
#define NROWT (NATOM/16)      /* 32  */
#define NCOLT (NCOL/16)       /* 864 */
#define NTILE (NROWT*NCOLT)   /* 27648 */
#define WAVES 8               /* waves (tiles) per 256-thread block */

/* workspace layout (float offsets) */
#define OFF_QX   0
#define OFF_QY   13824
#define OFF_QZ   27648
#define OFF_QN2  41472
#define OFF_PX   55296
#define OFF_PY   55808
#define OFF_PZ   56320
#define OFF_PN2  56832
#define OFF_CN   57344
#define OFF_ACC  57856
#define OFF_C6   57872
#define OFF_C8   (57872 + 262144)
/* total floats: 582160 -> ~2.33 MB of d_ws */

// ---------------------------------------------------------------------------
// prep: bohr-space positions, shifted images q = (pos[j] + shift_int[s]@cell)/AUTOANG,
// squared norms, zero cn + accumulator.
// ---------------------------------------------------------------------------
__global__ void prep_kernel(const float* __restrict__ pos,
                            const int* __restrict__ sint,
                            const float* __restrict__ cell,
                            float* __restrict__ w) {
  int n = blockIdx.x * blockDim.x + threadIdx.x;
  if (n < NCOL) {
    int s = n >> 9, j = n & 511;
    float s0 = (float)sint[s*3+0], s1 = (float)sint[s*3+1], s2 = (float)sint[s*3+2];
    float shx = s0*cell[0] + s1*cell[3] + s2*cell[6];
    float shy = s0*cell[1] + s1*cell[4] + s2*cell[7];
    float shz = s0*cell[2] + s1*cell[5] + s2*cell[8];
    float qx = (pos[j*3+0] + shx) * INV_AUTOANG;
    float qy = (pos[j*3+1] + shy) * INV_AUTOANG;
    float qz = (pos[j*3+2] + shz) * INV_AUTOANG;
    w[OFF_QX + n] = qx;  w[OFF_QY + n] = qy;  w[OFF_QZ + n] = qz;
    w[OFF_QN2 + n] = qx*qx + qy*qy + qz*qz;
  }
  if (n < NATOM) {
    float x = pos[n*3+0] * INV_AUTOANG;
    float y = pos[n*3+1] * INV_AUTOANG;
    float z = pos[n*3+2] * INV_AUTOANG;
    w[OFF_PX + n] = x;  w[OFF_PY + n] = y;  w[OFF_PZ + n] = z;
    w[OFF_PN2 + n] = x*x + y*y + z*z;
    w[OFF_CN + n] = 0.0f;
  }
  if (n == 0) w[OFF_ACC] = 0.0f;
}

// ---------------------------------------------------------------------------
// 16x16 tile of dot(pos_i, q_n) via V_WMMA_F32_16X16X4_F32.
// A (16x4): lanes 0-15 -> M=lane, VGPR0=K0(x), VGPR1=K1(y);
//           lanes 16-31 -> M=lane-16, VGPR0=K2(z), VGPR1=K3(0).
// B (4x16): lanes 0-15 -> N=lane, VGPR0=K0, VGPR1=K1;
//           lanes 16-31 -> N=lane-16, VGPR0=K2, VGPR1=K3.
// D (16x16 f32): VGPR v, lanes 0-15 -> row v; lanes 16-31 -> row v+8. Col = lane&15.
// ---------------------------------------------------------------------------
__device__ __forceinline__ v8f tile_dot(const float* __restrict__ w,
                                        int i0, int n0, int lane) {
  int  h  = lane & 15;
  bool hi = (lane & 16) != 0;
  v2f a, b;
  a.x = hi ? w[OFF_PZ + i0 + h] : w[OFF_PX + i0 + h];
  a.y = hi ? 0.0f               : w[OFF_PY + i0 + h];
  b.x = hi ? w[OFF_QZ + n0 + h] : w[OFF_QX + n0 + h];
  b.y = hi ? 0.0f               : w[OFF_QY + n0 + h];
  v8f c = {0.f,0.f,0.f,0.f,0.f,0.f,0.f,0.f};
  return __builtin_amdgcn_wmma_f32_16x16x4_f32(false, a, false, b,
                                               (short)0, c, false, false);
}

// ---------------------------------------------------------------------------
// CN pass: cn[i] += sigmoid damping over all in-range (j,s).
// ---------------------------------------------------------------------------
__global__ void cn_kernel(const int* __restrict__ Z,
                          const float* __restrict__ rcov,
                          const int* __restrict__ sint,
                          float* __restrict__ w) {
  int lane = threadIdx.x & 31;
  int tile = blockIdx.x * WAVES + (threadIdx.x >> 5);
  int ti = tile / NCOLT, tn = tile % NCOLT;
  int i0 = ti * 16, n0 = tn * 16;

  v8f dot = tile_dot(w, i0, n0, lane);

  int h   = lane & 15;
  int hi8 = (lane & 16) ? 8 : 0;
  int n = n0 + h, j = n & 511, s = n >> 9;
  float qn2 = w[OFF_QN2 + n];
  float rcj = rcov[Z[j]];
  bool  szero = ((sint[s*3] | sint[s*3+1] | sint[s*3+2]) == 0);

  float acc[8];
#pragma unroll
  for (int v = 0; v < 8; ++v) {
    int m = i0 + v + hi8;
    float r2 = w[OFF_PN2 + m] + qn2 - 2.0f * dot[v];
    bool ok = (r2 > 1e-8f) && (r2 <= CNTHR2) && !(szero && (j == m));
    float rinv = rsqrtf(fmaxf(r2, 1e-12f));
    float rco  = rcov[Z[m]] + rcj;
    float damp = 1.0f / (1.0f + __expf(-K1F * (rco * rinv - 1.0f)));
    acc[v] = ok ? damp : 0.0f;
  }
  // reduce across the 16 columns of each half-wave
#pragma unroll
  for (int mask = 1; mask < 16; mask <<= 1) {
#pragma unroll
    for (int v = 0; v < 8; ++v) acc[v] += __shfl_xor(acc[v], mask);
  }
  if (h == 0) {
#pragma unroll
    for (int v = 0; v < 8; ++v) atomicAdd(&w[OFF_CN + i0 + v + hi8], acc[v]);
  }
}

// ---------------------------------------------------------------------------
// C6 interpolation: one thread per (i,j). Two passes over the 25-point table.
// ---------------------------------------------------------------------------
__global__ void c6_kernel(const int* __restrict__ Z,
                          const float* __restrict__ c6ab,
                          const float* __restrict__ r2r4,
                          float* __restrict__ w) {
  int idx = blockIdx.x * blockDim.x + threadIdx.x;
  if (idx >= NATOM * NATOM) return;
  int i = idx >> 9, j = idx & 511;
  int zi = Z[i], zj = Z[j];
  const float* t = c6ab + (size_t)(zi * 95 + zj) * 75;   // [5][5][3]
  float cni = w[OFF_CN + i], cnj = w[OFF_CN + j];

  float dmin = 1e10f;
#pragma unroll
  for (int k = 0; k < 25; ++k) {
    float c6r = t[k*3], cna = t[k*3+1], cnb = t[k*3+2];
    float da = cni - cna, db = cnj - cnb;
    float dcn = da*da + db*db;
    dmin = (c6r > 0.0f) ? fminf(dmin, dcn) : dmin;
  }
  float num = 0.0f, den = 0.0f;
#pragma unroll
  for (int k = 0; k < 25; ++k) {
    float c6r = t[k*3], cna = t[k*3+1], cnb = t[k*3+2];
    float da = cni - cna, db = cnj - cnb;
    float wg = __expf(K3F * ((da*da + db*db) - dmin));
    wg = (c6r > 0.0f) ? wg : 0.0f;
    num += c6r * wg;
    den += wg;
  }
  float c6 = num / fmaxf(den, 1e-20f);
  w[OFF_C6 + idx] = c6;
  w[OFF_C8 + idx] = 3.0f * c6 * r2r4[zi] * r2r4[zj];
}

// ---------------------------------------------------------------------------
// Energy pass: e6 + e8 with multiply-chain powers, wave-reduced, one atomic/block-wave.
// ---------------------------------------------------------------------------
__global__ void energy_kernel(const int* __restrict__ Z,
                              const float* __restrict__ r0ab,
                              const int* __restrict__ sint,
                              float* __restrict__ w) {
  int lane = threadIdx.x & 31;
  int tile = blockIdx.x * WAVES + (threadIdx.x >> 5);
  int ti = tile / NCOLT, tn = tile % NCOLT;
  int i0 = ti * 16, n0 = tn * 16;

  v8f dot = tile_dot(w, i0, n0, lane);

  int h   = lane & 15;
  int hi8 = (lane & 16) ? 8 : 0;
  int n = n0 + h, j = n & 511, s = n >> 9;
  float qn2 = w[OFF_QN2 + n];
  int   zj  = Z[j];
  bool  szero = ((sint[s*3] | sint[s*3+1] | sint[s*3+2]) == 0);

  float esum = 0.0f;
#pragma unroll
  for (int v = 0; v < 8; ++v) {
    int m = i0 + v + hi8;
    float r2 = w[OFF_PN2 + m] + qn2 - 2.0f * dot[v];
    bool ok = (r2 > 1e-8f) && (r2 <= CUTOFF2) && !(szero && (j == m));
    float r2c   = fmaxf(r2, 1e-12f);
    float rinv2 = 1.0f / r2c;
    float rinv  = rsqrtf(r2c);
    float r0 = r0ab[Z[m] * 95 + zj];
    int pidx = m * NATOM + j;
    float c6 = w[OFF_C6 + pidx];
    float c8 = w[OFF_C8 + pidx];
    // t6 = (RS6*r0/r)^14, t8 = (r0/r)^16 — pure multiply chains
    float x = RS6 * r0 * rinv;
    float x2 = x*x, x4 = x2*x2, x8 = x4*x4;
    float t6 = x8 * x4 * x2;
    float y = r0 * rinv;
    float y2 = y*y, y4 = y2*y2, y8 = y4*y4;
    float t8 = y8 * y8;
    float r6inv = rinv2 * rinv2 * rinv2;
    float r8inv = r6inv * rinv2;
    float e = c6 * r6inv / (1.0f + 6.0f * t6)
            + S18 * c8 * r8inv / (1.0f + 6.0f * t8);
    esum += ok ? e : 0.0f;
  }
#pragma unroll
  for (int mask = 1; mask < 32; mask <<= 1) esum += __shfl_xor(esum, mask);
  if (lane == 0) atomicAdd(&w[OFF_ACC], esum);
}

__global__ void finalize_kernel(const float* __restrict__ w, float* __restrict__ out) {
  out[0] = -0.5f * AUTOEV * w[OFF_ACC];
}

// ---------------------------------------------------------------------------
extern "C" void kernel_launch(void* const* d_in, const int* in_sizes, int n_in,
                              void* d_out, int out_size, void* d_ws, size_t ws_size,
                              hipStream_t stream) {
  const int*   Z    = (const int*)  d_in[0];
  const float* pos  = (const float*)d_in[1];
  const int*   sint = (const int*)  d_in[2];
  const float* cell = (const float*)d_in[3];
  const float* c6ab = (const float*)d_in[4];
  const float* r0ab = (const float*)d_in[5];
  const float* rcov = (const float*)d_in[6];
  const float* r2r4 = (const float*)d_in[7];
  float* w   = (float*)d_ws;
  float* out = (float*)d_out;

  prep_kernel<<<(NCOL + 255) / 256, 256, 0, stream>>>(pos, sint, cell, w);
  cn_kernel<<<NTILE / WAVES, 32 * WAVES, 0, stream>>>(Z, rcov, sint, w);
  c6_kernel<<<(NATOM * NATOM) / 256, 256, 0, stream>>>(Z, c6ab, r2r4, w);
  energy_kernel<<<NTILE / WAVES, 32 * WAVES, 0, stream>>>(Z, r0ab, sint, w);
  finalize_kernel<<<1, 1, 0, stream>>>(w, out);
}